// GlobalGatedUpdate_49709951483915
// MI455X (gfx1250) — compile-verified
//
#include <hip/hip_runtime.h>

// Problem constants (match the reference)
#define V 49688
#define D 128
#define B 16
#define N 100

typedef float float4v __attribute__((ext_vector_type(4)));

// ---------------------------------------------------------------------------
// Kernel 1: broadcast-copy W into all B output slots.
// Each thread loads ONE float4 of W (regular temporal hint: W is re-used and
// L2-resident at 25.4 MB vs 192 MB L2) and issues B=16 non-temporal b128
// stores (TH=NT) so the 407 MB streaming output does not thrash L2.
// Grid: exactly V*D/4 / 256 = 6211 blocks of 256 threads (8 wave32 per block).
// ---------------------------------------------------------------------------
__global__ __launch_bounds__(256) void GGU_bcast_copy(
    const float4v* __restrict__ W4, float4v* __restrict__ out4) {
  const long long WN4 = (long long)V * (D / 4);  // 1,590,016 float4s per copy
  const long long widx = (long long)blockIdx.x * 256 + threadIdx.x;
  const float4v w = W4[widx];                    // RT load -> L2 keeps W hot
  float4v* p = out4 + widx;
#pragma unroll
  for (int b = 0; b < B; ++b) {
    __builtin_nontemporal_store(w, p);           // global_store_b128 th:NT
    p += WN4;
  }
}

// ---------------------------------------------------------------------------
// Kernel 2: scatter the gated rows.
// One wave32 per (b, i) row: 32 lanes x float4 = 128 floats = D exactly.
// out[b, node, :] = (1 - alpha[node]) * W[node, :] + alpha[node] * x[i, :]
// 1600 rows total -> 8 rows per 256-thread block -> 200 blocks.
// ---------------------------------------------------------------------------
__global__ __launch_bounds__(256) void GGU_scatter_update(
    const int* __restrict__ nodes,     // (B, N) row-major
    const float4v* __restrict__ x4,    // (B*N, D/4), only rows [0, N) used
    const float4v* __restrict__ W4,    // (V, D/4)
    const float* __restrict__ alpha,   // (V,)
    float4v* __restrict__ out4) {      // (B, V, D/4)
  const int lane = threadIdx.x & 31;
  const int row = blockIdx.x * 8 + (threadIdx.x >> 5);  // [0, B*N)
  if (row >= B * N) return;

  const int b = row / N;
  const int i = row - b * N;           // position within sample -> x row index
  const int node = nodes[row];
  const float a = alpha[node];
  const float one_m_a = 1.0f - a;

  const float4v w = W4[(long long)node * (D / 4) + lane];
  const float4v xv = x4[(long long)i * (D / 4) + lane];

  float4v r;
  r.x = one_m_a * w.x + a * xv.x;
  r.y = one_m_a * w.y + a * xv.y;
  r.z = one_m_a * w.z + a * xv.z;
  r.w = one_m_a * w.w + a * xv.w;

  out4[((long long)b * V + node) * (D / 4) + lane] = r;
}

extern "C" void kernel_launch(void* const* d_in, const int* in_sizes, int n_in,
                              void* d_out, int out_size, void* d_ws,
                              size_t ws_size, hipStream_t stream) {
  // setup_inputs() dict order: nodes (int32), x (f32), W (f32), alpha (f32)
  const int* nodes = (const int*)d_in[0];
  const float4v* x4 = (const float4v*)d_in[1];
  const float4v* W4 = (const float4v*)d_in[2];
  const float* alpha = (const float*)d_in[3];
  float4v* out4 = (float4v*)d_out;

  // V*D/4 = 1,590,016 is exactly divisible by 256 -> 6211 blocks, no tail.
  const int copy_blocks = (V * (D / 4)) / 256;  // 6211
  GGU_bcast_copy<<<copy_blocks, 256, 0, stream>>>(W4, out4);

  // 1600 rows, one wave32 per row, 8 rows per block -> 200 blocks.
  const int rows = B * N;
  GGU_scatter_update<<<rows / 8, 256, 0, stream>>>(nodes, x4, W4, alpha, out4);
}